// WHVILinear_34016140984791
// MI455X (gfx1250) — compile-verified
//
#include <hip/hip_runtime.h>

typedef __attribute__((ext_vector_type(2))) float v2f;
typedef __attribute__((ext_vector_type(8))) float v8f;

#define N_ROWS 8192
#define D_DIM  4096
#define ROWS_PER_WG 2
#define WG_THREADS (ROWS_PER_WG * 32)
#define ROW_PAD (D_DIM + D_DIM / 64)   // +1 float per 64 -> conflict-free strided stores

// Bank-conflict-avoiding LDS index mapping (64 banks x 4B).
__device__ __forceinline__ int lpad(int i) { return i + (i >> 6); }

// H16[p][q] = (-1)^popcount(p&q)
__device__ __forceinline__ float hsign16(int p, int q) {
    return (__builtin_popcount(p & q) & 1) ? -1.0f : 1.0f;
}

#define WMMA_F32X4(A, B, C) \
    __builtin_amdgcn_wmma_f32_16x16x4_f32(false, (A), false, (B), (short)0, (C), false, false)

// ---- Stage 1: S=1 axis, A-operands streamed from global x (non-temporal, fused * s2) ----
__device__ __forceinline__ void stage_in(const float* __restrict__ xrow,
                                         const float* __restrict__ s2,
                                         float* rbuf, int lane,
                                         v2f b0, v2f b1, v2f b2, v2f b3) {
    const int m = lane & 15, half = lane >> 4, kb = 2 * half;
    #pragma unroll 2
    for (int tile = 0; tile < 16; ++tile) {
        const int abase = m * 16 + tile * 256;
        v8f acc = {0.f, 0.f, 0.f, 0.f, 0.f, 0.f, 0.f, 0.f};
        #pragma unroll
        for (int c = 0; c < 4; ++c) {
            const int off = abase + kb + 4 * c;
            v2f xv = __builtin_nontemporal_load((const v2f*)(xrow + off));  // TH=NT stream
            v2f sv = *(const v2f*)(s2 + off);                               // hot, RT
            v2f a = xv * sv;
            acc = WMMA_F32X4(a, (c == 0 ? b0 : c == 1 ? b1 : c == 2 ? b2 : b3), acc);
        }
        #pragma unroll
        for (int r = 0; r < 8; ++r) {
            const int oidx = (r + 8 * half) * 16 + tile * 256 + m;  // lane-contiguous
            rbuf[lpad(oidx)] = acc[r];
        }
    }
}

// ---- Middle stages: LDS -> LDS; optionally fuse elementwise g on the store ----
template <int S, int LOWS, int HIGHS, bool GMUL>
__device__ __forceinline__ void stage_mid(float* rbuf, const float* __restrict__ g,
                                          int lane, v2f b0, v2f b1, v2f b2, v2f b3) {
    const int m = lane & 15, half = lane >> 4, kb = 2 * half;
    #pragma unroll 2
    for (int tile = 0; tile < 16; ++tile) {
        const int abase = m * LOWS + tile * HIGHS;
        v8f acc = {0.f, 0.f, 0.f, 0.f, 0.f, 0.f, 0.f, 0.f};
        v2f a;
        a.x = rbuf[lpad(abase + (kb + 0) * S)];  a.y = rbuf[lpad(abase + (kb + 1) * S)];
        acc = WMMA_F32X4(a, b0, acc);
        a.x = rbuf[lpad(abase + (kb + 4) * S)];  a.y = rbuf[lpad(abase + (kb + 5) * S)];
        acc = WMMA_F32X4(a, b1, acc);
        a.x = rbuf[lpad(abase + (kb + 8) * S)];  a.y = rbuf[lpad(abase + (kb + 9) * S)];
        acc = WMMA_F32X4(a, b2, acc);
        a.x = rbuf[lpad(abase + (kb + 12) * S)]; a.y = rbuf[lpad(abase + (kb + 13) * S)];
        acc = WMMA_F32X4(a, b3, acc);
        #pragma unroll
        for (int r = 0; r < 8; ++r) {
            const int oidx = (r + 8 * half) * LOWS + tile * HIGHS + m * S;
            float v = acc[r];
            if (GMUL) v *= g[oidx];          // g is 16 KB, L0/L2-resident
            rbuf[lpad(oidx)] = v;            // padded -> conflict-free banks
        }
    }
}

// ---- Stage 6: S=1 axis from LDS, fused * s1, non-temporal coalesced store to HBM ----
__device__ __forceinline__ void stage_out(const float* rbuf,
                                          const float* __restrict__ s1,
                                          float* __restrict__ orow, int lane,
                                          v2f b0, v2f b1, v2f b2, v2f b3) {
    const int m = lane & 15, half = lane >> 4, kb = 2 * half;
    #pragma unroll 2
    for (int tile = 0; tile < 16; ++tile) {
        const int abase = m * 16 + tile * 256;
        v8f acc = {0.f, 0.f, 0.f, 0.f, 0.f, 0.f, 0.f, 0.f};
        v2f a;
        a.x = rbuf[lpad(abase + kb + 0)];  a.y = rbuf[lpad(abase + kb + 1)];
        acc = WMMA_F32X4(a, b0, acc);
        a.x = rbuf[lpad(abase + kb + 4)];  a.y = rbuf[lpad(abase + kb + 5)];
        acc = WMMA_F32X4(a, b1, acc);
        a.x = rbuf[lpad(abase + kb + 8)];  a.y = rbuf[lpad(abase + kb + 9)];
        acc = WMMA_F32X4(a, b2, acc);
        a.x = rbuf[lpad(abase + kb + 12)]; a.y = rbuf[lpad(abase + kb + 13)];
        acc = WMMA_F32X4(a, b3, acc);
        #pragma unroll
        for (int r = 0; r < 8; ++r) {
            const int oidx = (r + 8 * half) * 16 + tile * 256 + m;  // lane-contiguous 64B halves
            __builtin_nontemporal_store(acc[r] * s1[oidx], orow + oidx);  // TH=NT stream
        }
    }
}

// Fold g = g_mu + softplus(g_rho) * epsilon once into workspace.
__global__ void whvi_gprep_kernel(const float* __restrict__ g_mu,
                                  const float* __restrict__ g_rho,
                                  const float* __restrict__ eps,
                                  float* __restrict__ g) {
    int i = blockIdx.x * blockDim.x + threadIdx.x;
    if (i < D_DIM) {
        float r  = g_rho[i];
        float sp = fmaxf(r, 0.0f) + log1pf(__expf(-fabsf(r)));  // stable softplus
        g[i] = g_mu[i] + sp * eps[i];
    }
}

__global__ __launch_bounds__(WG_THREADS)
void whvi_main_kernel(const float* __restrict__ x,
                      const float* __restrict__ s1,
                      const float* __restrict__ s2,
                      const float* __restrict__ g,
                      float* __restrict__ out) {
    __shared__ float lds[ROWS_PER_WG * ROW_PAD];   // ~32.5 KB; one padded row per wave

    const int wave = threadIdx.x >> 5;
    const int lane = threadIdx.x & 31;
    const int row  = blockIdx.x * ROWS_PER_WG + wave;

    float*       rbuf = lds + wave * ROW_PAD;
    const float* xrow = x   + (size_t)row * D_DIM;
    float*       orow = out + (size_t)row * D_DIM;

    // Constant B operands: H16 K-chunks (4x16, 2 VGPRs), built from popcount parity.
    const int n    = lane & 15;
    const int half = lane >> 4;
    v2f b0, b1, b2, b3;
    b0.x = hsign16(0  + 2 * half, n); b0.y = hsign16(1  + 2 * half, n);
    b1.x = hsign16(4  + 2 * half, n); b1.y = hsign16(5  + 2 * half, n);
    b2.x = hsign16(8  + 2 * half, n); b2.y = hsign16(9  + 2 * half, n);
    b3.x = hsign16(12 + 2 * half, n); b3.y = hsign16(13 + 2 * half, n);

    // FWHT #1 (axes k, j, i) — stage 1 streams x*s2 from global.
    stage_in(xrow, s2, rbuf, lane, b0, b1, b2, b3);
    stage_mid<16, 1, 256, false>(rbuf, g, lane, b0, b1, b2, b3);
    stage_mid<256, 1, 16, true >(rbuf, g, lane, b0, b1, b2, b3);   // fused t *= g

    // FWHT #2 (axes i, j, k) — reversed order so the last stage is S=1.
    stage_mid<256, 1, 16, false>(rbuf, g, lane, b0, b1, b2, b3);
    stage_mid<16, 1, 256, false>(rbuf, g, lane, b0, b1, b2, b3);
    stage_out(rbuf, s1, orow, lane, b0, b1, b2, b3);               // fused * s1, store to HBM
}

extern "C" void kernel_launch(void* const* d_in, const int* in_sizes, int n_in,
                              void* d_out, int out_size, void* d_ws, size_t ws_size,
                              hipStream_t stream) {
    (void)in_sizes; (void)n_in; (void)out_size; (void)ws_size;

    const float* x    = (const float*)d_in[0];
    const float* s1   = (const float*)d_in[1];
    const float* s2   = (const float*)d_in[2];
    const float* g_mu = (const float*)d_in[3];
    const float* g_rho= (const float*)d_in[4];
    const float* eps  = (const float*)d_in[5];
    float* out = (float*)d_out;
    float* g   = (float*)d_ws;   // D_DIM floats of scratch

    hipLaunchKernelGGL(whvi_gprep_kernel, dim3(D_DIM / 256), dim3(256), 0, stream,
                       g_mu, g_rho, eps, g);
    hipLaunchKernelGGL(whvi_main_kernel, dim3(N_ROWS / ROWS_PER_WG), dim3(WG_THREADS), 0, stream,
                       x, s1, s2, g, out);
}